// _ScaledDotProductAttention_684
// MI455X (gfx1250) — compile-verified
//
#include <hip/hip_runtime.h>

#define BS   2
#define NH   16
#define SEQ  2048
#define DH   64
#define QT   (SEQ / 16)   // q-row tiles per (b,h)
#define CT   (SEQ / 16)   // column tiles per row

typedef float v2f __attribute__((ext_vector_type(2)));
typedef float v8f __attribute__((ext_vector_type(8)));

// Pass 1: scores = (Q @ K) * scale + prev ; write scores; lane-local online
// softmax stats, combined across the 16-lane row group once at the end.
__global__ void __launch_bounds__(32)
attn_scores_kernel(const float* __restrict__ q,
                   const float* __restrict__ k,     // (b,h,D,S) pre-transposed
                   const float* __restrict__ prev,
                   const float* __restrict__ scale_p,
                   float* __restrict__ scores,
                   float* __restrict__ mbuf,
                   float* __restrict__ lbuf) {
    const int lane = threadIdx.x;
    const int lid  = lane & 15;
    const int hh   = lane >> 4;            // lane half: 0 or 1
    const int bh   = blockIdx.x / QT;
    const int qt   = blockIdx.x % QT;
    const float scale = scale_p[0];

    const float* qb = q    + (size_t)bh * SEQ * DH + (size_t)qt * 16 * DH;
    const float* kb = k    + (size_t)bh * DH * SEQ;
    const float* pb = prev + (size_t)bh * SEQ * SEQ + (size_t)qt * 16 * SEQ;
    float*       sb = scores + (size_t)bh * SEQ * SEQ + (size_t)qt * 16 * SEQ;

    // A-fragments of the 16x64 Q tile: lane lid row, K pair (4i + 2*hh, +1).
    v2f aq[16];
    #pragma unroll
    for (int i = 0; i < 16; ++i)
        aq[i] = *(const v2f*)(qb + lid * DH + i * 4 + hh * 2);

    // Lane-private running stats: this lane only ever sees column s0+lid.
    float m_ln[8], l_ln[8];
    #pragma unroll
    for (int r = 0; r < 8; ++r) { m_ln[r] = -3.0e38f; l_ln[r] = 0.0f; }

    for (int st = 0; st < CT; ++st) {
        const int s0 = st * 16;
        v8f acc = {};
        // B-fragment: K rows (d) strided by SEQ, contiguous in s.
        const float* kp = kb + (size_t)(hh * 2) * SEQ + s0 + lid;
        #pragma unroll
        for (int i = 0; i < 16; ++i) {
            v2f bk;
            bk.x = kp[(size_t)(4 * i)     * SEQ];
            bk.y = kp[(size_t)(4 * i + 1) * SEQ];
            acc = __builtin_amdgcn_wmma_f32_16x16x4_f32(
                false, aq[i], false, bk, (short)0, acc, false, false);
        }
        const float* pr = pb + (size_t)(hh * 8) * SEQ + s0 + lid;
        float*       sr = sb + (size_t)(hh * 8) * SEQ + s0 + lid;
        #pragma unroll
        for (int r = 0; r < 8; ++r) {
            float s = acc[r] * scale + pr[(size_t)r * SEQ];
            sr[(size_t)r * SEQ] = s;
            // Lane-local online update (no cross-lane traffic in hot loop).
            float mn = fmaxf(m_ln[r], s);
            l_ln[r]  = l_ln[r] * __expf(m_ln[r] - mn) + __expf(s - mn);
            m_ln[r]  = mn;
        }
    }

    // One associative (m,l) combine across the 16-lane row group.
    #pragma unroll
    for (int off = 8; off; off >>= 1) {
        #pragma unroll
        for (int r = 0; r < 8; ++r) {
            float mo = __shfl_xor(m_ln[r], off, 32);
            float lo = __shfl_xor(l_ln[r], off, 32);
            float mn = fmaxf(m_ln[r], mo);
            l_ln[r]  = l_ln[r] * __expf(m_ln[r] - mn) + lo * __expf(mo - mn);
            m_ln[r]  = mn;
        }
    }

    if (lid == 0) {
        const int base = bh * SEQ + qt * 16 + hh * 8;
        #pragma unroll
        for (int r = 0; r < 8; ++r) {
            mbuf[base + r] = m_ln[r];
            lbuf[base + r] = l_ln[r];
        }
    }
}

// Pass 2: weights = exp(scores - m) / l ; write weights ; out = W @ V.
__global__ void __launch_bounds__(32)
attn_out_kernel(const float* __restrict__ v,       // (b,h,S,D)
                const float* __restrict__ scores,
                const float* __restrict__ mbuf,
                const float* __restrict__ lbuf,
                float* __restrict__ weights,
                float* __restrict__ outp) {
    __shared__ float wt[16 * 18];                  // padded 16x16 W tile
    const int lane = threadIdx.x;
    const int lid  = lane & 15;
    const int hh   = lane >> 4;
    const int bh   = blockIdx.x / QT;
    const int qt   = blockIdx.x % QT;

    const float* vb = v      + (size_t)bh * SEQ * DH;
    const float* sb = scores + (size_t)bh * SEQ * SEQ + (size_t)qt * 16 * SEQ;
    float*       wb = weights + (size_t)bh * SEQ * SEQ + (size_t)qt * 16 * SEQ;
    float*       ob = outp   + (size_t)bh * SEQ * DH + (size_t)qt * 16 * DH;

    float mr[8], rl[8];
    const int sbase = bh * SEQ + qt * 16 + hh * 8;
    #pragma unroll
    for (int r = 0; r < 8; ++r) {
        mr[r] = mbuf[sbase + r];
        rl[r] = 1.0f / lbuf[sbase + r];
    }

    v8f o0 = {}, o1 = {}, o2 = {}, o3 = {};       // 16x64 output accumulators

    for (int st = 0; st < CT; ++st) {
        const int s0 = st * 16;
        const float* sr = sb + (size_t)(hh * 8) * SEQ + s0 + lid;
        float*       wr = wb + (size_t)(hh * 8) * SEQ + s0 + lid;
        #pragma unroll
        for (int r = 0; r < 8; ++r) {
            float w = __expf(sr[(size_t)r * SEQ] - mr[r]) * rl[r];
            wr[(size_t)r * SEQ] = w;
            wt[(r + hh * 8) * 18 + lid] = w;      // stage for transpose
        }
        __syncthreads();                           // single-wave WG: cheap fence
        const float* vp = vb + (size_t)(s0 + hh * 2) * DH + lid;
        #pragma unroll
        for (int i = 0; i < 4; ++i) {              // K steps within tile
            v2f aw = *(const v2f*)(&wt[lid * 18 + i * 4 + hh * 2]);
            const float* vpi = vp + (size_t)(4 * i) * DH;
            v2f b0, b1, b2, b3;
            b0.x = vpi[0];  b0.y = vpi[DH + 0];
            b1.x = vpi[16]; b1.y = vpi[DH + 16];
            b2.x = vpi[32]; b2.y = vpi[DH + 32];
            b3.x = vpi[48]; b3.y = vpi[DH + 48];
            o0 = __builtin_amdgcn_wmma_f32_16x16x4_f32(false, aw, false, b0, (short)0, o0, false, false);
            o1 = __builtin_amdgcn_wmma_f32_16x16x4_f32(false, aw, false, b1, (short)0, o1, false, false);
            o2 = __builtin_amdgcn_wmma_f32_16x16x4_f32(false, aw, false, b2, (short)0, o2, false, false);
            o3 = __builtin_amdgcn_wmma_f32_16x16x4_f32(false, aw, false, b3, (short)0, o3, false, false);
        }
        __syncthreads();                           // protect wt before overwrite
    }
    #pragma unroll
    for (int r = 0; r < 8; ++r) {
        const int row = r + hh * 8;
        ob[row * DH +  0 + lid] = o0[r];
        ob[row * DH + 16 + lid] = o1[r];
        ob[row * DH + 32 + lid] = o2[r];
        ob[row * DH + 48 + lid] = o3[r];
    }
}

extern "C" void kernel_launch(void* const* d_in, const int* in_sizes, int n_in,
                              void* d_out, int out_size, void* d_ws, size_t ws_size,
                              hipStream_t stream) {
    const float* q     = (const float*)d_in[0];
    const float* k     = (const float*)d_in[1];
    const float* v     = (const float*)d_in[2];
    const float* prev  = (const float*)d_in[3];
    const float* scale = (const float*)d_in[4];

    float* out   = (float*)d_out;
    float* out_o = out;                                        // (BS,H,S,D)
    float* out_w = out_o + (size_t)BS * NH * SEQ * DH;         // (BS,H,S,S)
    float* out_s = out_w + (size_t)BS * NH * SEQ * SEQ;        // (BS,H,S,S)

    float* mbuf = (float*)d_ws;                                // BS*H*S floats
    float* lbuf = mbuf + (size_t)BS * NH * SEQ;                // BS*H*S floats

    dim3 grid(BS * NH * QT);
    dim3 block(32);
    attn_scores_kernel<<<grid, block, 0, stream>>>(q, k, prev, scale, out_s, mbuf, lbuf);
    attn_out_kernel<<<grid, block, 0, stream>>>(v, out_s, mbuf, lbuf, out_w, out_o);
}